// Seq2Seq_32401233281558
// MI455X (gfx1250) — compile-verified
//
#include <hip/hip_runtime.h>
#include <hip/hip_bf16.h>
#include <math.h>

// Problem constants (from reference)
#define I_DIM 1739
#define H_DIM 256
#define G3    (3 * H_DIM)   // 768
#define T_DIM 2048
#define KC    100           // k-slices of W_hh^T cached in LDS (100*768*4 = 300KB)

typedef float v2f __attribute__((ext_vector_type(2)));
typedef float v8f __attribute__((ext_vector_type(8)));

// ---------------------------------------------------------------------------
// WMMA fp32 GEMM:  C[M][N] = A[M][K] * W[N][K]^T + bias[N]
// One wave (32 threads) computes a 32x16 output tile (two 16x16 M-tiles
// sharing one B fragment) with V_WMMA_F32_16X16X4_F32:
//   - halves B traffic vs one tile/wave (~5.3 FLOP/B from L2)
//   - two independent accumulator chains -> no WMMA->WMMA RAW on D
// M must be a multiple of 32; N,K arbitrary.
//
// N-tail: clamp (not mask) the B column — garbage B columns only affect
// D columns we never store => all loads unconditional, no exec juggling.
// K-tail: clamp the address, zero the value (v_cndmask, no branch).
//
// Fragment layouts per CDNA5 ISA (CDNA5 ISA):
//  A (16x4 f32, 2 VGPRs): lane<16 holds A[lane][0..1], lane>=16 holds A[lane-16][2..3]
//  B (4x16 f32, 2 VGPRs): lane<16 holds B[0..1][lane], lane>=16 holds B[2..3][lane-16]
//  C/D (16x16 f32, 8 VGPRs): vgpr r -> row r (lanes 0-15) / row r+8 (lanes 16-31), col = lane%16
// ---------------------------------------------------------------------------
__global__ __launch_bounds__(32)
void k_gemm_wmma_f32(const float* __restrict__ A,     // [M][K]
                     const float* __restrict__ W,     // [N][K]
                     const float* __restrict__ bias,  // [N]
                     float* __restrict__ C,           // [M][N]
                     int M, int N, int K)
{
    const int m0   = blockIdx.x * 32;
    const int n0   = blockIdx.y * 16;
    const int lane = threadIdx.x;       // 0..31
    const int half = lane >> 4;         // 0 or 1
    const int l16  = lane & 15;

    const int  ncol  = n0 + l16;
    const bool colok = ncol < N;
    const int  ncl   = colok ? ncol : (N - 1);       // clamped column

    const float* __restrict__ arow0 = A + (size_t)(m0 + l16) * K;
    const float* __restrict__ arow1 = A + (size_t)(m0 + 16 + l16) * K;
    const float* __restrict__ wcol  = W + (size_t)ncl * K;

    v8f acc0 = {}, acc1 = {};
    const int kfull = K & ~3;
    int k = 0;
    #pragma unroll 2
    for (; k < kfull; k += 4) {
        const int ka = k + 2 * half;
        v2f b, a0, a1;
        b[0]  = wcol[ka];      b[1]  = wcol[ka + 1];
        a0[0] = arow0[ka];     a0[1] = arow0[ka + 1];
        a1[0] = arow1[ka];     a1[1] = arow1[ka + 1];
        acc0 = __builtin_amdgcn_wmma_f32_16x16x4_f32(
                   false, a0, false, b, (short)0, acc0, false, false);
        acc1 = __builtin_amdgcn_wmma_f32_16x16x4_f32(
                   false, a1, false, b, (short)0, acc1, false, false);
    }
    if (k < K) {  // K-tail (K=1739 -> 3 leftover): clamp address, zero value
        const int  ka  = k + 2 * half;
        const int  kc  = K - 1;
        const int  k0c = ka     < K ? ka     : kc;
        const int  k1c = ka + 1 < K ? ka + 1 : kc;
        const bool ok0 = ka     < K;
        const bool ok1 = ka + 1 < K;
        v2f b, a0, a1;
        b[0]  = ok0 ? wcol[k0c]  : 0.0f;  b[1]  = ok1 ? wcol[k1c]  : 0.0f;
        a0[0] = ok0 ? arow0[k0c] : 0.0f;  a0[1] = ok1 ? arow0[k1c] : 0.0f;
        a1[0] = ok0 ? arow1[k0c] : 0.0f;  a1[1] = ok1 ? arow1[k1c] : 0.0f;
        acc0 = __builtin_amdgcn_wmma_f32_16x16x4_f32(
                   false, a0, false, b, (short)0, acc0, false, false);
        acc1 = __builtin_amdgcn_wmma_f32_16x16x4_f32(
                   false, a1, false, b, (short)0, acc1, false, false);
    }

    if (colok) {
        const float bv = bias ? bias[ncol] : 0.0f;
        #pragma unroll
        for (int r = 0; r < 8; ++r) {
            const int row = m0 + r + 8 * half;
            C[(size_t)row * N + ncol]        = acc0[r] + bv;
            C[(size_t)(row + 16) * N + ncol] = acc1[r] + bv;
        }
    }
}

// ---------------------------------------------------------------------------
// Transpose W[rows][cols] -> Wt[cols][rows]  (W_hh: 768x256 -> 256x768)
// Makes the recurrence's weight reads lane-contiguous (coalesced).
// ---------------------------------------------------------------------------
__global__ void k_transpose(const float* __restrict__ W,
                            float* __restrict__ Wt,
                            int rows, int cols)
{
    const int idx = blockIdx.x * blockDim.x + threadIdx.x;
    if (idx < rows * cols) {
        const int r = idx / cols;
        const int c = idx % cols;
        Wt[(size_t)c * rows + r] = W[idx];
    }
}

// ---------------------------------------------------------------------------
// GRU recurrence: one persistent workgroup of 768 threads (24 wave32 waves).
// CDNA5 gives 320KB LDS per WGP: cache the first KC=100 k-slices of W_hh^T
// in LDS (300KB, loaded once, amortized over 2048 steps); stream the
// remaining 156 slices (468KB/step, L2-resident) — cuts the L2 traffic of
// the bandwidth-bound scan by ~1.7x.  LDS reads are bank-conflict free
// (lane-contiguous Wlds[k*768+t]); h is read via float4 LDS broadcasts.
// Dynamic LDS: 311,296 B requested at launch (< 320KB WGP limit).
// ---------------------------------------------------------------------------
__global__ __launch_bounds__(G3)
void k_gru(const float* __restrict__ xp,    // [T][768]  (x @ W_ih^T + b_ih)
           const float* __restrict__ Wt,    // [256][768], Wt[k][t] = W_hh[t][k]
           const float* __restrict__ b_hh,  // [768]
           const float* __restrict__ h0,    // [256] or nullptr (zeros)
           float* __restrict__ hs,          // [T][256] or nullptr
           float* __restrict__ hT,          // [256] or nullptr
           int T)
{
    extern __shared__ float smem[];
    float* __restrict__ Wlds = smem;                 // [KC][768]  300KB
    float* __restrict__ h    = smem + KC * G3;       // [256]  (16B aligned)
    float* __restrict__ hh   = h + H_DIM;            // [768]

    const int t = threadIdx.x;

    // Preload LDS weight cache (coalesced, once)
    for (int k = 0; k < KC; ++k)
        Wlds[k * G3 + t] = Wt[(size_t)k * G3 + t];

    if (t < H_DIM) h[t] = h0 ? h0[t] : 0.0f;
    const float bias = b_hh[t];
    __syncthreads();

    for (int step = 0; step < T; ++step) {
        // hh[t] = dot(h, W_hh[t,:]) + b_hh[t]
        float acc = bias;
        // LDS-cached k-range [0, KC)
        #pragma unroll 4
        for (int k = 0; k < KC; k += 4) {
            const float4 hv = *(const float4*)&h[k];
            acc = fmaf(hv.x, Wlds[(k + 0) * G3 + t], acc);
            acc = fmaf(hv.y, Wlds[(k + 1) * G3 + t], acc);
            acc = fmaf(hv.z, Wlds[(k + 2) * G3 + t], acc);
            acc = fmaf(hv.w, Wlds[(k + 3) * G3 + t], acc);
        }
        // L2-streamed k-range [KC, 256)
        const float* __restrict__ wp = Wt + (size_t)KC * G3 + t;
        #pragma unroll 4
        for (int k = KC; k < H_DIM; k += 4) {
            const float4 hv = *(const float4*)&h[k];
            acc = fmaf(hv.x, wp[(size_t)(k - KC + 0) * G3], acc);
            acc = fmaf(hv.y, wp[(size_t)(k - KC + 1) * G3], acc);
            acc = fmaf(hv.z, wp[(size_t)(k - KC + 2) * G3], acc);
            acc = fmaf(hv.w, wp[(size_t)(k - KC + 3) * G3], acc);
        }
        hh[t] = acc;
        __syncthreads();

        if (t < H_DIM) {
            const float* __restrict__ xrow = xp + (size_t)step * G3;
            const float r  = 1.0f / (1.0f + expf(-(xrow[t]             + hh[t])));
            const float z  = 1.0f / (1.0f + expf(-(xrow[H_DIM + t]     + hh[H_DIM + t])));
            const float n  = tanhf(xrow[2 * H_DIM + t] + r * hh[2 * H_DIM + t]);
            const float hn = (1.0f - z) * n + z * h[t];
            h[t] = hn;
            if (hs) hs[(size_t)step * H_DIM + t] = hn;
        }
        __syncthreads();
    }

    if (t < H_DIM && hT) hT[t] = h[t];
}

// ---------------------------------------------------------------------------
// Launch: xproj GEMMs (WMMA) -> W_hh transposes -> enc scan -> dec scan
//         -> output projection (WMMA) into d_out.
// ---------------------------------------------------------------------------
extern "C" void kernel_launch(void* const* d_in, const int* in_sizes, int n_in,
                              void* d_out, int out_size, void* d_ws, size_t ws_size,
                              hipStream_t stream)
{
    const float* x        = (const float*)d_in[0];   // [1][2048][1739]
    const float* W_ih_enc = (const float*)d_in[2];   // [768][1739]
    const float* W_hh_enc = (const float*)d_in[3];   // [768][256]
    const float* b_ih_enc = (const float*)d_in[4];   // [768]
    const float* b_hh_enc = (const float*)d_in[5];   // [768]
    const float* W_ih_dec = (const float*)d_in[6];
    const float* W_hh_dec = (const float*)d_in[7];
    const float* b_ih_dec = (const float*)d_in[8];
    const float* b_hh_dec = (const float*)d_in[9];
    const float* W_out    = (const float*)d_in[10];  // [1739][256]
    const float* b_out    = (const float*)d_in[11];  // [1739]
    float*       out      = (float*)d_out;           // [2048][1][1739]

    // Workspace layout (floats), ~16.3 MB total
    float* ws     = (float*)d_ws;
    float* xp_enc = ws;                              // 2048*768
    float* xp_dec = xp_enc + (size_t)T_DIM * G3;     // 2048*768
    float* Wt_enc = xp_dec + (size_t)T_DIM * G3;     // 256*768
    float* Wt_dec = Wt_enc + (size_t)H_DIM * G3;     // 256*768
    float* h_enc  = Wt_dec + (size_t)H_DIM * G3;     // 256
    float* hs_dec = h_enc + H_DIM;                   // 2048*256

    // 1) Input projections for both GRUs (independent, WMMA f32)
    dim3 gXP(T_DIM / 32, G3 / 16);                   // 64 x 48 tiles (32x16 each)
    k_gemm_wmma_f32<<<gXP, 32, 0, stream>>>(x, W_ih_enc, b_ih_enc, xp_enc,
                                            T_DIM, G3, I_DIM);
    k_gemm_wmma_f32<<<gXP, 32, 0, stream>>>(x, W_ih_dec, b_ih_dec, xp_dec,
                                            T_DIM, G3, I_DIM);

    // 2) Transpose recurrent weights for coalesced scan reads
    const int nW = G3 * H_DIM;
    k_transpose<<<(nW + 255) / 256, 256, 0, stream>>>(W_hh_enc, Wt_enc, G3, H_DIM);
    k_transpose<<<(nW + 255) / 256, 256, 0, stream>>>(W_hh_dec, Wt_dec, G3, H_DIM);

    // 3) Sequential scans (single persistent workgroup each, 311KB dynamic LDS)
    const size_t gruLds = (size_t)(KC * G3 + H_DIM + G3) * sizeof(float);
    k_gru<<<1, G3, gruLds, stream>>>(xp_enc, Wt_enc, b_hh_enc,
                                     nullptr, nullptr, h_enc, T_DIM);
    k_gru<<<1, G3, gruLds, stream>>>(xp_dec, Wt_dec, b_hh_dec,
                                     h_enc, hs_dec, nullptr, T_DIM);

    // 4) Output projection preds = hs_dec @ W_out^T + b_out (WMMA f32)
    dim3 gOP(T_DIM / 32, (I_DIM + 15) / 16);         // 64 x 109 tiles
    k_gemm_wmma_f32<<<gOP, 32, 0, stream>>>(hs_dec, W_out, b_out, out,
                                            T_DIM, I_DIM, H_DIM);
}